// MLE_new_42339787604431
// MI455X (gfx1250) — compile-verified
//
#include <hip/hip_runtime.h>
#include <cmath>

// Wave32-per-emitter LM-MLE fitter for 16x16 Gaussian ROIs.
// mu = inten * (gyv ⊗ gxv) + bg computed as ONE v_wmma_f32_16x16x4_f32
// rank-1 outer product per emitter per iteration (separable Gaussian:
// only 2 exps per lane instead of 8).

typedef __attribute__((ext_vector_type(2))) float v2f;
typedef __attribute__((ext_vector_type(8))) float v8f;

#if __has_builtin(__builtin_amdgcn_wmma_f32_16x16x4_f32)
#define MLE_USE_WMMA 1
#else
#define MLE_USE_WMMA 0
#endif

__device__ __forceinline__ float wsum32(float v) {
#pragma unroll
  for (int off = 16; off > 0; off >>= 1) v += __shfl_xor(v, off, 32);
  return v;
}

__global__ __launch_bounds__(256) void mle_lm_kernel(
    const float* __restrict__ initial, const float* __restrict__ smp,
    const float* __restrict__ prmm, const int* __restrict__ itersPtr,
    const float* __restrict__ lamPtr, float* __restrict__ out, int B) {
  const int lane = threadIdx.x & 31;
  const int wv = threadIdx.x >> 5;
  const int b = blockIdx.x * (blockDim.x >> 5) + wv;
  if (b >= B) return;  // wave-uniform: EXEC stays all-ones for WMMA

  const int iters = *itersPtr;
  const float lam = *lamPtr;

  const float SIG = 1.3f;
  const float inv2s2 = 1.0f / (2.0f * SIG * SIG);
  const float invs2 = 1.0f / (SIG * SIG);

  float* outCur = out;                  // (B,4)
  float* outLL = out + (size_t)B * 4;   // (B,)
  float* outTr = out + (size_t)B * 5;   // (iters+1,B,4)

  // parameters (wave-uniform)
  float cur0 = initial[(size_t)b * 4 + 0];
  float cur1 = initial[(size_t)b * 4 + 1];
  float cur2 = initial[(size_t)b * 4 + 2];
  float cur3 = initial[(size_t)b * 4 + 3];
  const float mn0 = prmm[0], mx0 = prmm[1];
  const float mn1 = prmm[2], mx1 = prmm[3];
  const float mn2 = prmm[4], mx2 = prmm[5];
  const float mn3 = prmm[6], mx3 = prmm[7];

  // this lane's 8 pixels: column x=lane&15, rows y0..y0+7 (WMMA C/D layout)
  const int col = lane & 15;
  const int y0 = (lane >> 4) * 8;
  float s[8];
  const float* sp = smp + (size_t)b * 256;
#pragma unroll
  for (int r = 0; r < 8; ++r) s[r] = sp[(y0 + r) * 16 + col];

  // traces[0] = initial
  if (lane < 4) {
    float v = lane == 0 ? cur0 : lane == 1 ? cur1 : lane == 2 ? cur2 : cur3;
    outTr[(size_t)b * 4 + lane] = v;
  }

  float scl0 = 0.f, scl1 = 0.f, scl2 = 0.f, scl3 = 0.f;
  float tA0 = 0.f, tA1 = 0.f, tA2 = 0.f, tA3 = 0.f;            // traces[i-1]
  float tB0 = cur0, tB1 = cur1, tB2 = cur2, tB3 = cur3;        // traces[i]
  bool ba0 = true, ba1 = true, ba2 = true, ba3 = true;
  bool good = true;
  float ms[8];
#pragma unroll
  for (int r = 0; r < 8; ++r) ms[r] = 0.f;

  for (int i = 0; i < iters; ++i) {
    // ---- separable Gaussian model ----
    const float m = (float)(lane & 15);
    const float dx = m - cur0;  // gx for this lane's column (col == lane&15)
    const float dy = m - cur1;
    const float gxm = __expf(-(dx * dx) * inv2s2);  // gxv[col]
    const float gym = __expf(-(dy * dy) * inv2s2);  // gyv[lane&15]

    float muA[8];
#if MLE_USE_WMMA
    // D(16x16) = A(16x4) x B(4x16) + C ; only K=0 nonzero:
    // A[m,0] = inten*gyv[m]  (VGPR0, lanes 0-15) ; B[0,n] = gxv[n] (VGPR0, lanes 0-15)
    v2f a;
    a.x = (lane < 16) ? (cur2 * gym) : 0.0f;
    a.y = 0.0f;
    v2f bm;
    bm.x = (lane < 16) ? gxm : 0.0f;
    bm.y = 0.0f;
    v8f c = {cur3, cur3, cur3, cur3, cur3, cur3, cur3, cur3};
    v8f d = __builtin_amdgcn_wmma_f32_16x16x4_f32(false, a, false, bm,
                                                  (short)0, c, false, false);
#pragma unroll
    for (int r = 0; r < 8; ++r) muA[r] = d[r];
#else
    const float tI = cur2 * gxm;
#pragma unroll
    for (int r = 0; r < 8; ++r) {
      float gyr = __shfl(gym, y0 + r, 32);  // gyv[y0+r] lives in lane y0+r
      muA[r] = fmaf(tI, gyr, cur3);
    }
#endif

    // ---- per-pixel Hessian/gradient accumulation ----
    float A00 = 0, A01 = 0, A02 = 0, A03 = 0, A11 = 0, A12 = 0, A13 = 0,
          A22 = 0, A23 = 0, A33 = 0;
    float g0 = 0, g1 = 0, g2 = 0, g3 = 0;
    const float rint = __builtin_amdgcn_rcpf(cur2);  // inten >= 1 (clamped)
#pragma unroll
    for (int r = 0; r < 8; ++r) {
      const float mu = muA[r];
      if (good) ms[r] = mu;  // mu_store gating (uniform)
      const float gyr = (float)(y0 + r) - cur1;
      const float muc = fmaxf(mu, 1e-9f);
      const float im = __builtin_amdgcn_rcpf(muc);
      const float af = s[r] * im * im;
      const float w = fmaf(s[r], im, -1.0f);
      const float q = mu - cur3;  // inten * g
      const float t = q * invs2;
      const float j0 = t * dx;
      const float j1 = t * gyr;
      const float j2 = q * rint;  // g ; j3 = 1
      const float a0 = af * j0, a1 = af * j1, a2 = af * j2;
      A00 = fmaf(a0, j0, A00); A01 = fmaf(a0, j1, A01);
      A02 = fmaf(a0, j2, A02); A03 += a0;
      A11 = fmaf(a1, j1, A11); A12 = fmaf(a1, j2, A12); A13 += a1;
      A22 = fmaf(a2, j2, A22); A23 += a2; A33 += af;
      g0 = fmaf(w, j0, g0); g1 = fmaf(w, j1, g1);
      g2 = fmaf(w, j2, g2); g3 += w;
    }
    // ---- butterfly reduce 14 sums across the wave ----
    A00 = wsum32(A00); A01 = wsum32(A01); A02 = wsum32(A02); A03 = wsum32(A03);
    A11 = wsum32(A11); A12 = wsum32(A12); A13 = wsum32(A13);
    A22 = wsum32(A22); A23 = wsum32(A23); A33 = wsum32(A33);
    g0 = wsum32(g0); g1 = wsum32(g1); g2 = wsum32(g2); g3 = wsum32(g3);

    // scale = max(col-norm^2 of undamped alpha, scale_old)
    const float c0 = A00 * A00 + A01 * A01 + A02 * A02 + A03 * A03;
    const float c1 = A01 * A01 + A11 * A11 + A12 * A12 + A13 * A13;
    const float c2 = A02 * A02 + A12 * A12 + A22 * A22 + A23 * A23;
    const float c3 = A03 * A03 + A13 * A13 + A23 * A23 + A33 * A33;
    const float ns0 = fmaxf(c0, scl0), ns1 = fmaxf(c1, scl1);
    const float ns2 = fmaxf(c2, scl2), ns3 = fmaxf(c3, scl3);

    // ---- solve (alpha + lam*scale*I) steps = beta ; SPD, no pivot ----
    float M[4][5] = {{A00 + lam * ns0, A01, A02, A03, g0},
                     {A01, A11 + lam * ns1, A12, A13, g1},
                     {A02, A12, A22 + lam * ns2, A23, g2},
                     {A03, A13, A23, A33 + lam * ns3, g3}};
#pragma unroll
    for (int k = 0; k < 4; ++k) {
      const float piv = 1.0f / M[k][k];
#pragma unroll
      for (int rr = k + 1; rr < 4; ++rr) {
        const float f = M[rr][k] * piv;
#pragma unroll
        for (int cc = k; cc < 5; ++cc) M[rr][cc] -= f * M[k][cc];
      }
    }
    const float st3 = M[3][4] / M[3][3];
    const float st2 = (M[2][4] - M[2][3] * st3) / M[2][2];
    const float st1 = (M[1][4] - M[1][2] * st2 - M[1][3] * st3) / M[1][1];
    const float st0 =
        (M[0][4] - M[0][1] * st1 - M[0][2] * st2 - M[0][3] * st3) / M[0][0];

    const float n0 = fminf(fmaxf(cur0 + st0, mn0), mx0);
    const float n1 = fminf(fmaxf(cur1 + st1, mn1), mx1);
    const float n2 = fminf(fmaxf(cur2 + st2, mn2), mx2);
    const float n3 = fminf(fmaxf(cur3 + st3, mn3), mx3);

    if (good) {
      cur0 = n0; cur1 = n1; cur2 = n2; cur3 = n3;
      scl0 = ns0; scl1 = ns1; scl2 = ns2; scl3 = ns3;
    }
    const float tn0 = good ? cur0 : 0.f, tn1 = good ? cur1 : 0.f;
    const float tn2 = good ? cur2 : 0.f, tn3 = good ? cur3 : 0.f;
    if (lane < 4) {
      float v = lane == 0 ? tn0 : lane == 1 ? tn1 : lane == 2 ? tn2 : tn3;
      outTr[((size_t)(i + 1) * B + b) * 4 + lane] = v;
    }
    // convergence: delta = |traces[i-1] - traces[i]|
    ba0 = good ? (fabsf(tA0 - tB0) < 0.1f) : ba0;
    ba1 = good ? (fabsf(tA1 - tB1) < 0.1f) : ba1;
    ba2 = good ? (fabsf(tA2 - tB2) < 0.1f) : ba2;
    ba3 = good ? (fabsf(tA3 - tB3) < 0.1f) : ba3;
    good = !(ba0 && ba1 && ba2 && ba3);
    tA0 = tB0; tA1 = tB1; tA2 = tB2; tA3 = tB3;
    tB0 = tn0; tB1 = tn1; tB2 = tn2; tB3 = tn3;

    if (!good) {  // converged: remaining trace rows stay zero; stop early
      for (int j = i + 2; j <= iters; ++j)
        if (lane < 4) outTr[((size_t)j * B + b) * 4 + lane] = 0.0f;
      break;
    }
  }

  // loglik = sum(smp*log(mu_store/smp)) - sum(mu_store - smp)
  float ll = 0.0f;
#pragma unroll
  for (int r = 0; r < 8; ++r)
    ll += s[r] * __logf(ms[r] / s[r]) - (ms[r] - s[r]);
  ll = wsum32(ll);
  if (isinf(ll)) ll = 1e-20f;
  if (lane == 0) outLL[b] = ll;
  if (lane < 4) {
    float v = lane == 0 ? cur0 : lane == 1 ? cur1 : lane == 2 ? cur2 : cur3;
    outCur[(size_t)b * 4 + lane] = v;
  }
}

extern "C" void kernel_launch(void* const* d_in, const int* in_sizes, int n_in,
                              void* d_out, int out_size, void* d_ws,
                              size_t ws_size, hipStream_t stream) {
  const float* initial = (const float*)d_in[0];
  const float* smp = (const float*)d_in[1];
  const float* prmm = (const float*)d_in[2];
  const int* iters = (const int*)d_in[3];
  const float* lam = (const float*)d_in[4];
  float* out = (float*)d_out;
  const int B = in_sizes[0] / 4;
  const int wavesPerBlock = 8;  // 256 threads = 8 wave32 = 8 emitters
  const int blocks = (B + wavesPerBlock - 1) / wavesPerBlock;
  mle_lm_kernel<<<blocks, wavesPerBlock * 32, 0, stream>>>(initial, smp, prmm,
                                                           iters, lam, out, B);
}